// PortOpt_DL_DeepSig_35055523070800
// MI455X (gfx1250) — compile-verified
//
#include <hip/hip_runtime.h>
#include <hip/hip_bf16.h>

typedef float v2f __attribute__((ext_vector_type(2)));
typedef float v8f __attribute__((ext_vector_type(8)));

#define WAVES_PER_BLOCK 4
#define NSTEP 56         // padded steps (14 groups of 4)
#define SIG2N 121
#define SIGCH 1463       // 11 + 121 + 1331
#define NSTOCK 10

__device__ __forceinline__ float relu_f(float v) { return v > 0.0f ? v : 0.0f; }

__global__ __launch_bounds__(128) void deepsig_kernel(
    const float* __restrict__ x,    // (B, 64, 8)
    const float* __restrict__ W1,   // (8,8,4)
    const float* __restrict__ b1,   // (8)
    const float* __restrict__ W2,   // (8,8,4)
    const float* __restrict__ b2,   // (8)
    const float* __restrict__ W3,   // (2,8,4)
    const float* __restrict__ b3,   // (2)
    const float* __restrict__ Wl,   // (10, 1463)
    const float* __restrict__ bl,   // (10)
    float* __restrict__ out,        // (B, 10)
    int B)
{
    __shared__ float s_W1[256], s_W2[256], s_W3[64];
    __shared__ float s_b1[8], s_b2[8], s_b3[2];
    __shared__ float __align__(16) s_x [WAVES_PER_BLOCK][512];   // 64*8
    __shared__ float s_h1[WAVES_PER_BLOCK][488];   // 8*61
    __shared__ float s_h2[WAVES_PER_BLOCK][464];   // 8*58
    __shared__ float s_a [WAVES_PER_BLOCK][605];   // 55*11
    __shared__ float s_dx[WAVES_PER_BLOCK][616];   // 56*11 (last step zero)
    __shared__ float s_s1[WAVES_PER_BLOCK][16];    // sig1 (11 used)
    __shared__ float s_s2[WAVES_PER_BLOCK][128];   // sig2 (121 used)
    __shared__ float s_vf[WAVES_PER_BLOCK][16];    // per-step vfac = sig1/2 + dx/6
    __shared__ float s_uf[WAVES_PER_BLOCK][16];    // per-step ufac = sig1 + dx/2

    const int tid  = threadIdx.x;
    const int w    = tid >> 5;
    const int lane = tid & 31;
    const int L    = lane & 15;
    const int hi   = lane >> 4;

    // ---- load conv weights to LDS (workgroup-wide) ----
    for (int i = tid; i < 256; i += 128) { s_W1[i] = W1[i]; s_W2[i] = W2[i]; }
    for (int i = tid; i < 64;  i += 128) s_W3[i] = W3[i];
    if (tid < 8) { s_b1[tid] = b1[tid]; s_b2[tid] = b2[tid]; }
    if (tid < 2) s_b3[tid] = b3[tid];
    __syncthreads();

    const int b = blockIdx.x * WAVES_PER_BLOCK + w;
    if (b >= B) return;   // after the only workgroup-wide barrier

    float* xw  = s_x [w];
    float* h1w = s_h1[w];
    float* h2w = s_h2[w];
    float* aw  = s_a [w];
    float* dxw = s_dx[w];
    float* s1w = s_s1[w];
    float* s2w = s_s2[w];
    float* vfw = s_vf[w];
    float* ufw = s_uf[w];

    // ================= Phase 1: augmentation =================
    // stage x[b] (512 floats) into LDS, 128-bit coalesced
    {
        const float4* xb4 = (const float4*)(x + (size_t)b * 512);
        float4* xw4 = (float4*)xw;
        #pragma unroll
        for (int i = 0; i < 4; ++i) xw4[lane + 32 * i] = xb4[lane + 32 * i];
    }
    // conv1: h1[o,t] = b1[o] + sum_{i,k} W1[o,i,k] * x[t+k, i],  t=0..60
    for (int idx = lane; idx < 488; idx += 32) {
        int o = idx / 61, t = idx % 61;
        float acc = s_b1[o];
        #pragma unroll
        for (int i = 0; i < 8; ++i)
            #pragma unroll
            for (int k = 0; k < 4; ++k)
                acc += s_W1[(o * 8 + i) * 4 + k] * xw[(t + k) * 8 + i];
        h1w[o * 61 + t] = acc;
    }
    // conv2 on relu(h1): t=0..57
    for (int idx = lane; idx < 464; idx += 32) {
        int o = idx / 58, t = idx % 58;
        float acc = s_b2[o];
        #pragma unroll
        for (int i = 0; i < 8; ++i)
            #pragma unroll
            for (int k = 0; k < 4; ++k)
                acc += s_W2[(o * 8 + i) * 4 + k] * relu_f(h1w[i * 61 + (t + k)]);
        h2w[o * 58 + t] = acc;
    }
    // assemble augmented path a[t,c], t=0..54:
    //   c<8: orig x[t+9,c]; c==8: t/54; c>=9: conv3 on relu(h2)
    for (int idx = lane; idx < 605; idx += 32) {
        int t = idx / 11, c = idx % 11;
        float val;
        if (c < 8) {
            val = xw[(t + 9) * 8 + c];
        } else if (c == 8) {
            val = (float)t * (1.0f / 54.0f);
        } else {
            int o = c - 9;
            float acc = s_b3[o];
            #pragma unroll
            for (int i = 0; i < 8; ++i)
                #pragma unroll
                for (int k = 0; k < 4; ++k)
                    acc += s_W3[(o * 8 + i) * 4 + k] * relu_f(h2w[i * 58 + (t + k)]);
            val = acc;
        }
        aw[idx] = val;
    }
    // increments: dx[0]=a[0]; dx[t]=a[t]-a[t-1]; dx[55]=0 (padding step)
    for (int idx = lane; idx < 616; idx += 32) {
        int t = idx / 11;
        float v;
        if (t == 0)      v = aw[idx];
        else if (t < 55) v = aw[idx] - aw[idx - 11];
        else             v = 0.0f;
        dxw[idx] = v;
    }

    // ================= Phase 2: signature scan =================
    // init sig1 / sig2 in LDS
    if (lane < 16) s1w[lane] = 0.0f;
    #pragma unroll
    for (int e = 0; e < 4; ++e) s2w[lane + 32 * e] = 0.0f;

    // per-lane precomputed indices
    int pm[8], qm[8];        // for T rows:  row = 16m + L
    #pragma unroll
    for (int m = 0; m < 8; ++m) { int r = 16 * m + L; pm[m] = r / 11; qm[m] = r % 11; }
    int pe[4], qe[4];        // for owned sig2 entries: e = lane + 32j
    #pragma unroll
    for (int e = 0; e < 4; ++e) { int r = lane + 32 * e; pe[e] = r / 11; qe[e] = r % 11; }

    v8f cacc[8];
    #pragma unroll
    for (int m = 0; m < 8; ++m) cacc[m] = (v8f){0,0,0,0,0,0,0,0};

    for (int g = 0; g < NSTEP / 4; ++g) {
        const int t0 = 4 * g;
        float aK0[8], aK1[8];   // A-matrix lane slices (VGPR0 / VGPR1)
        #pragma unroll
        for (int j = 0; j < 4; ++j) {
            const int t = t0 + j;
            const float* dxt = &dxw[t * 11];
            // per-channel step factors from OLD sig1; also advance sig1
            if (lane < 11) {
                float d  = dxt[lane];
                float s1 = s1w[lane];
                vfw[lane] = 0.5f * s1 + (1.0f / 6.0f) * d;
                ufw[lane] = s1 + 0.5f * d;
                s1w[lane] = s1 + d;
            }
            // A build: at step j, lanes with hi == j>>1 capture T (uses OLD sig2)
            // T_pq = sig2_pq + (sig1_p/2 + dx_p/6) * dx_q
            if (hi == (j >> 1)) {
                #pragma unroll
                for (int m = 0; m < 8; ++m) {
                    int row = 16 * m + L;
                    float val = 0.0f;
                    if (row < SIG2N)
                        val = s2w[row] + vfw[pm[m]] * dxt[qm[m]];
                    if ((j & 1) == 0) aK0[m] = val; else aK1[m] = val;
                }
            }
            // sig2 update (OLD sig1 via ufac): sig2_pq += (sig1_p + dx_p/2) * dx_q
            #pragma unroll
            for (int e = 0; e < 4; ++e) {
                int idx = lane + 32 * e;
                if (idx < SIG2N)
                    s2w[idx] += ufw[pe[e]] * dxt[qe[e]];
            }
        }
        // B matrix: lane holds N=L, K = v + 2*hi  ->  dx[t0 + v + 2*hi][L]
        v2f bV;
        bV[0] = (L < 11) ? dxw[(t0 + 2 * hi + 0) * 11 + L] : 0.0f;
        bV[1] = (L < 11) ? dxw[(t0 + 2 * hi + 1) * 11 + L] : 0.0f;
        // sig3 += T(121x4) @ DX(4x11) via 8 row tiles of V_WMMA_F32_16X16X4_F32
        #pragma unroll
        for (int m = 0; m < 8; ++m) {
            v2f aV; aV[0] = aK0[m]; aV[1] = aK1[m];
            cacc[m] = __builtin_amdgcn_wmma_f32_16x16x4_f32(
                false, aV, false, bV, (short)0, cacc[m], false, false);
        }
    }

    // ================= Phase 3: projection sig @ Wl^T + bl =================
    float acc[NSTOCK];
    #pragma unroll
    for (int s = 0; s < NSTOCK; ++s) acc[s] = 0.0f;

    // sig3 lives in C tiles: element c[m][v] = sig3[pq = 16m + v + 8*hi, r = L]
    if (L < 11) {
        #pragma unroll
        for (int m = 0; m < 8; ++m) {
            #pragma unroll
            for (int v = 0; v < 8; ++v) {
                int pq = 16 * m + v + 8 * hi;
                if (pq < SIG2N) {
                    float val = cacc[m][v];
                    int j = 132 + pq * 11 + L;
                    #pragma unroll
                    for (int s = 0; s < NSTOCK; ++s)
                        acc[s] += val * Wl[s * SIGCH + j];
                }
            }
        }
    }
    // sig1 (j=0..10) and sig2 (j=11..131) contributions, lane-distributed
    for (int jj = lane; jj < 132; jj += 32) {
        float val = (jj < 11) ? s1w[jj] : s2w[jj - 11];
        #pragma unroll
        for (int s = 0; s < NSTOCK; ++s)
            acc[s] += val * Wl[s * SIGCH + jj];
    }
    // wave32 reduction + write
    #pragma unroll
    for (int s = 0; s < NSTOCK; ++s) {
        float v = acc[s];
        v += __shfl_xor(v, 16, 32);
        v += __shfl_xor(v, 8, 32);
        v += __shfl_xor(v, 4, 32);
        v += __shfl_xor(v, 2, 32);
        v += __shfl_xor(v, 1, 32);
        if (lane == 0) out[(size_t)b * NSTOCK + s] = v + bl[s];
    }
}

extern "C" void kernel_launch(void* const* d_in, const int* in_sizes, int n_in,
                              void* d_out, int out_size, void* d_ws, size_t ws_size,
                              hipStream_t stream) {
    (void)n_in; (void)out_size; (void)d_ws; (void)ws_size;
    const float* x  = (const float*)d_in[0];
    const float* W1 = (const float*)d_in[1];
    const float* b1 = (const float*)d_in[2];
    const float* W2 = (const float*)d_in[3];
    const float* b2 = (const float*)d_in[4];
    const float* W3 = (const float*)d_in[5];
    const float* b3 = (const float*)d_in[6];
    const float* Wl = (const float*)d_in[7];
    const float* bl = (const float*)d_in[8];
    float* out = (float*)d_out;

    const int B = in_sizes[0] / (64 * 8);
    const int grid = (B + WAVES_PER_BLOCK - 1) / WAVES_PER_BLOCK;
    deepsig_kernel<<<grid, 128, 0, stream>>>(x, W1, b1, W2, b2, W3, b3, Wl, bl, out, B);
}